// FFDGenerator_11682311045578
// MI455X (gfx1250) — compile-verified
//
#include <hip/hip_runtime.h>
#include <hip/hip_bf16.h>

// FFD cubic-B-spline upsampling: mesh (4,3,23,27,23) f32 -> flow (4,3,160,192,160) f32.
// Memory-bound (236 MB of stores ~ 10us @ 23.3 TB/s). Strategy:
//  - block = (bc, x-tile of 16, 8 consecutive y), 8 waves; contiguous 3105-float mesh
//    slab (5 cx planes) bulk-loaded to LDS,
//  - per-wave y-contraction into s[5][23] in LDS,
//  - per z-tile: z-contraction into the B operand, then the x-axis 4-tap contraction
//    done with two V_WMMA_F32_16X16X4_F32 (pivot-split 16x5 weight band),
//  - D stored with N=z in lanes -> coalesced b32 stores along the contiguous axis.

typedef float v2f __attribute__((ext_vector_type(2)));
typedef float v8f __attribute__((ext_vector_type(8)));

#define CPX 23
#define CPY 27
#define CPZ 23
#define PLANE (CPY * CPZ)          // 621
#define MESH_BC (CPX * PLANE)      // 14283
#define XS 160
#define YS 192
#define ZS 160
#define OUT_BC (XS * YS * ZS)      // 4915200
#define SLAB (5 * PLANE)           // 3105 floats of mesh per block
#define SPAD 120                   // s region per wave (115 used, padded)

__device__ __forceinline__ void bspline_w(float d, float& w0, float& w1,
                                          float& w2, float& w3) {
  float d2 = d * d;
  float d3 = d2 * d;
  float om = 1.0f - d;
  w0 = om * om * om * (1.0f / 6.0f);
  w1 = 0.5f * d3 - d2 + (2.0f / 3.0f);
  w2 = -0.5f * d3 + 0.5f * d2 + 0.5f * d + (1.0f / 6.0f);
  w3 = d3 * (1.0f / 6.0f);
}

__global__ __launch_bounds__(256) void ffd_flow_kernel(
    const float* __restrict__ mesh, float* __restrict__ out) {
  __shared__ float lds[SLAB + 8 * SPAD];

  const int tid = threadIdx.x;
  const int tx = blockIdx.x;        // x tile: x0 = 16*tx, cx base = 2*tx
  const int yg = blockIdx.y;        // y octet: y0 = 8*yg, qy = yg
  const int bc = blockIdx.z;        // b*3+c

  // ---- bulk load contiguous mesh slab: cx in [2tx, 2tx+4], all (cy,cz) ----
  const float* gsrc = mesh + bc * MESH_BC + (2 * tx) * PLANE;
  for (int i = tid; i < SLAB; i += 256) lds[i] = gsrc[i];
  __syncthreads();

  const int wave = tid >> 5;
  const int lane = tid & 31;
  const int y = 8 * yg + wave;      // this wave's y; ry = wave, qy = yg

  // ---- per-wave y contraction: s[c][cz] = sum_j wY[j] * m[c][qy+j][cz] ----
  float wy0, wy1, wy2, wy3;
  bspline_w((float)(y & 7) * 0.125f, wy0, wy1, wy2, wy3);
  float* sW = lds + SLAB + wave * SPAD;
  {
    const int qy = y >> 3;          // == yg
    for (int i = lane; i < 5 * CPZ; i += 32) {
      int c = i / CPZ;
      int cz = i - c * CPZ;
      const float* mb = lds + c * PLANE + qy * CPZ + cz;
      sW[i] = wy0 * mb[0] + wy1 * mb[CPZ] + wy2 * mb[2 * CPZ] + wy3 * mb[3 * CPZ];
    }
  }
  __syncthreads();

  // ---- per-lane constant A matrices (x-axis weights, pivot-split) --------
  // A layout (16x4 f32, M x K): lanes 0-15 hold K={0,1} in {.x,.y}; lanes
  // 16-31 hold K={2,3}. M = lane&15. Rows 0-7 use cx base 2tx (A0), rows
  // 8-15 use cx base 2tx+1 (A1); the inactive half of each A is zero.
  const int laneM = lane & 15;
  const bool hi = lane >= 16;
  float ax0, ax1, ax2, ax3;
  bspline_w((float)(laneM & 7) * 0.125f, ax0, ax1, ax2, ax3);
  const float klo = hi ? ax2 : ax0;
  const float khi = hi ? ax3 : ax1;
  v2f A0, A1;
  A0.x = (laneM < 8) ? klo : 0.0f;
  A0.y = (laneM < 8) ? khi : 0.0f;
  A1.x = (laneM >= 8) ? klo : 0.0f;
  A1.y = (laneM >= 8) ? khi : 0.0f;

  // ---- per-lane z weights (z local position is fixed per lane) -----------
  float wz0, wz1, wz2, wz3;
  bspline_w((float)(laneM & 7) * 0.125f, wz0, wz1, wz2, wz3);
  const int cbase = hi ? 2 : 0;     // local cx tap base for B operands

  // Output addressing: idx = bc*OUT_BC + x*(YS*ZS) + y*ZS + z
  const int xrow = 16 * tx + (hi ? 8 : 0);
  float* obase = out + (size_t)bc * OUT_BC + (size_t)xrow * (YS * ZS) +
                 (size_t)y * ZS + laneM;

  for (int zt = 0; zt < 10; ++zt) {
    const int qz = 2 * zt + (laneM >> 3);   // pivot for z = 16*zt + laneM
    const float* s0 = sW + cbase * CPZ + qz;

    // z-contraction for the 3 cx taps this lane feeds into B0/B1
    float ta = wz0 * s0[0] + wz1 * s0[1] + wz2 * s0[2] + wz3 * s0[3];
    const float* s1 = s0 + CPZ;
    float tb = wz0 * s1[0] + wz1 * s1[1] + wz2 * s1[2] + wz3 * s1[3];
    const float* s2 = s1 + CPZ;
    float tc = wz0 * s2[0] + wz1 * s2[1] + wz2 * s2[2] + wz3 * s2[3];

    // B layout (4x16 f32, K x N): lanes 0-15 N=lane hold K={0,1} in {.x,.y};
    // lanes 16-31 hold K={2,3}. B1 is the +1-shifted cx band.
    v2f B0, B1;
    B0.x = ta; B0.y = tb;
    B1.x = tb; B1.y = tc;

    v8f acc = {};
    acc = __builtin_amdgcn_wmma_f32_16x16x4_f32(false, A0, false, B0,
                                                (short)0, acc, false, false);
    acc = __builtin_amdgcn_wmma_f32_16x16x4_f32(false, A1, false, B1,
                                                (short)0, acc, false, false);

    // D: VGPR v holds x-row (xrow+v), lane gives z = 16*zt + laneM.
    float* o = obase + zt * 16;
#pragma unroll
    for (int v = 0; v < 8; ++v) {
      o[(size_t)v * (YS * ZS)] = acc[v];
    }
  }
}

extern "C" void kernel_launch(void* const* d_in, const int* in_sizes, int n_in,
                              void* d_out, int out_size, void* d_ws, size_t ws_size,
                              hipStream_t stream) {
  const float* mesh = (const float*)d_in[0];
  float* out = (float*)d_out;
  dim3 grid(XS / 16, YS / 8, 12);   // (10, 24, 12)
  dim3 block(256);
  ffd_flow_kernel<<<grid, block, 0, stream>>>(mesh, out);
}